// Mamba_81260781240888
// MI455X (gfx1250) — compile-verified
//
#include <hip/hip_runtime.h>
#include <math.h>

// ---- model constants ----
#define BATCH   2
#define SEQ     1024
#define D_MODEL 1024
#define D_INNER 2048
#define D_STATE 16
#define DT_RANK 64
#define D_CONV  4
#define M_ROWS  (BATCH * SEQ)   // 2048

typedef float v2f __attribute__((ext_vector_type(2)));
typedef float v8f __attribute__((ext_vector_type(8)));

static __device__ __forceinline__ v8f wmma_f32_k4(v2f a, v2f b, v8f c) {
    // D = A(16x4,f32) * B(4x16,f32) + C(16x16,f32)
    return __builtin_amdgcn_wmma_f32_16x16x4_f32(
        /*neg_a=*/false, a, /*neg_b=*/false, b,
        /*c_mod=*/(short)0, c, /*reuse_a=*/false, /*reuse_b=*/false);
}

static __device__ __forceinline__ float siluf(float x) {
    return x / (1.0f + __expf(-x));
}
static __device__ __forceinline__ float softplusf(float x) {
    return (x > 20.0f) ? x : log1pf(__expf(x));
}

// ---------------------------------------------------------------------------
// RMSNorm: one block per row (B*S rows), D_MODEL columns
// ---------------------------------------------------------------------------
__global__ void mamba_rmsnorm(const float* __restrict__ x,
                              const float* __restrict__ w,
                              float* __restrict__ out, int d) {
    __shared__ float red[256];
    const int row = blockIdx.x;
    const int tid = threadIdx.x;
    const float* xr = x + (size_t)row * d;
    float s = 0.0f;
    for (int i = tid; i < d; i += 256) { float v = xr[i]; s += v * v; }
    red[tid] = s;
    __syncthreads();
    for (int off = 128; off > 0; off >>= 1) {
        if (tid < off) red[tid] += red[tid + off];
        __syncthreads();
    }
    const float scale = rsqrtf(red[0] / (float)d + 1e-5f);
    float* orow = out + (size_t)row * d;
    for (int i = tid; i < d; i += 256) orow[i] = xr[i] * scale * w[i];
}

// ---------------------------------------------------------------------------
// Tiled WMMA GEMM:  C[m,n] = act( sum_k A[m,k]*B[n,k] + bias[n] ) + add[m,n]
// A: MxK (lda), B: NxK (ldb), C: MxN (ldc).  Requires M%64==0, N%64==0, K%32==0.
// act: 0 = none, 1 = softplus. bias/add may be null.
// Block: 256 threads (8 waves), 64x64 output tile, KC=32.
// ---------------------------------------------------------------------------
__global__ __launch_bounds__(256)
void mamba_gemm_nt_64x64(const float* __restrict__ A, const float* __restrict__ B,
                         const float* __restrict__ bias, const float* __restrict__ add,
                         float* __restrict__ C,
                         int K, int lda, int ldb, int ldc, int act) {
    __shared__ float As[64][33];
    __shared__ float Bs[64][33];

    const int tid  = threadIdx.x;
    const int lane = tid & 31;
    const int wave = tid >> 5;
    const int m0 = blockIdx.y * 64;
    const int n0 = blockIdx.x * 64;

    const int tm  = wave & 3;        // 0..3
    const int tn0 = wave >> 2;       // 0..1 ; second subtile at tn0+2

    v8f acc0 = {0.f,0.f,0.f,0.f,0.f,0.f,0.f,0.f};
    v8f acc1 = {0.f,0.f,0.f,0.f,0.f,0.f,0.f,0.f};

    const int row  = tm * 16 + (lane & 15);
    const int col0 = tn0 * 16 + (lane & 15);
    const int col1 = col0 + 32;
    const int kh   = (lane >> 4) << 1;   // 0 or 2

    for (int k0 = 0; k0 < K; k0 += 32) {
        // cooperative load: 64x32 floats per matrix = 512 float4, 2 per thread
        #pragma unroll
        for (int l = 0; l < 2; ++l) {
            const int idx = tid + l * 256;
            const int r   = idx >> 3;
            const int c4  = (idx & 7) << 2;
            const float4 av = *(const float4*)&A[(size_t)(m0 + r) * lda + k0 + c4];
            As[r][c4 + 0] = av.x; As[r][c4 + 1] = av.y;
            As[r][c4 + 2] = av.z; As[r][c4 + 3] = av.w;
            const float4 bv = *(const float4*)&B[(size_t)(n0 + r) * ldb + k0 + c4];
            Bs[r][c4 + 0] = bv.x; Bs[r][c4 + 1] = bv.y;
            Bs[r][c4 + 2] = bv.z; Bs[r][c4 + 3] = bv.w;
        }
        __syncthreads();
        #pragma unroll
        for (int kk = 0; kk < 32; kk += 4) {
            v2f a;  a.x  = As[row][kk + kh];  a.y  = As[row][kk + kh + 1];
            v2f b0; b0.x = Bs[col0][kk + kh]; b0.y = Bs[col0][kk + kh + 1];
            v2f b1; b1.x = Bs[col1][kk + kh]; b1.y = Bs[col1][kk + kh + 1];
            acc0 = wmma_f32_k4(a, b0, acc0);
            acc1 = wmma_f32_k4(a, b1, acc1);
        }
        __syncthreads();
    }

    // epilogue; C/D layout: vgpr r -> M = r + 8*(lane>=16), N = lane&15
    const int mbase = m0 + tm * 16 + ((lane >> 4) << 3);
    const int na = n0 + tn0 * 16 + (lane & 15);
    const int nb = na + 32;
    const float ba = bias ? bias[na] : 0.0f;
    const float bb = bias ? bias[nb] : 0.0f;
    #pragma unroll
    for (int r = 0; r < 8; ++r) {
        const int m = mbase + r;
        float va = acc0[r] + ba;
        float vb = acc1[r] + bb;
        if (act == 1) { va = softplusf(va); vb = softplusf(vb); }
        if (add) {
            va += add[(size_t)m * ldc + na];
            vb += add[(size_t)m * ldc + nb];
        }
        C[(size_t)m * ldc + na] = va;
        C[(size_t)m * ldc + nb] = vb;
    }
}

// ---------------------------------------------------------------------------
// Per-wave direct WMMA GEMM (for skinny N, e.g. x_proj N=96).
// C[m,n] = sum_k A[m,k]*B[n,k]. Requires M%16==0, N%16==0, K%4==0.
// ---------------------------------------------------------------------------
__global__ __launch_bounds__(128)
void mamba_gemm_nt_wave(const float* __restrict__ A, const float* __restrict__ B,
                        float* __restrict__ C,
                        int M, int N, int K, int lda, int ldb, int ldc) {
    const int gtid = blockIdx.x * blockDim.x + threadIdx.x;
    const int wid  = gtid >> 5;
    const int lane = gtid & 31;
    const int tilesN = N >> 4;
    const int total  = (M >> 4) * tilesN;
    if (wid < total) {
        const int m0 = (wid / tilesN) << 4;
        const int n0 = (wid % tilesN) << 4;
        const int kh = (lane >> 4) << 1;
        const float* ap = &A[(size_t)(m0 + (lane & 15)) * lda + kh];
        const float* bp = &B[(size_t)(n0 + (lane & 15)) * ldb + kh];
        v8f acc = {0.f,0.f,0.f,0.f,0.f,0.f,0.f,0.f};
        for (int k = 0; k < K; k += 4) {
            const v2f a = *(const v2f*)(ap + k);
            const v2f b = *(const v2f*)(bp + k);
            acc = wmma_f32_k4(a, b, acc);
        }
        const int mbase = m0 + ((lane >> 4) << 3);
        const int n = n0 + (lane & 15);
        #pragma unroll
        for (int r = 0; r < 8; ++r)
            C[(size_t)(mbase + r) * ldc + n] = acc[r];
    }
}

// ---------------------------------------------------------------------------
// Causal depthwise conv (width 4) + bias + SiLU.
// xz: (B*S, 2*D_INNER) — conv over the first D_INNER cols. out: (B*S, D_INNER)
// ---------------------------------------------------------------------------
__global__ void mamba_conv_silu(const float* __restrict__ xz,
                                const float* __restrict__ cw,
                                const float* __restrict__ cb,
                                float* __restrict__ out) {
    const int total = BATCH * SEQ * D_INNER;
    for (int idx = blockIdx.x * blockDim.x + threadIdx.x; idx < total;
         idx += gridDim.x * blockDim.x) {
        const int c = idx % D_INNER;
        const int t = (idx / D_INNER) % SEQ;
        const int b = idx / (D_INNER * SEQ);
        float acc = cb[c];
        #pragma unroll
        for (int k = 0; k < D_CONV; ++k) {
            const int tt = t - (D_CONV - 1) + k;
            if (tt >= 0)
                acc += xz[((size_t)(b * SEQ + tt)) * (2 * D_INNER) + c] * cw[c * D_CONV + k];
        }
        out[(size_t)(b * SEQ + t) * D_INNER + c] = siluf(acc);
    }
}

// ---------------------------------------------------------------------------
// Selective scan. One thread per (batch, channel). B/C (shared across
// channels) staged into LDS in 64-timestep chunks.
// dbc: (B*S, 96): [0:64]=delta_rank (unused here), [64:80]=Bm, [80:96]=Cm
// delta: (B*S, D_INNER) post-softplus. xc: (B*S, D_INNER). y: (B*S, D_INNER)
// grid: (D_INNER/256, BATCH), block: 256
// ---------------------------------------------------------------------------
__global__ __launch_bounds__(256)
void mamba_scan(const float* __restrict__ dbc,
                const float* __restrict__ delta,
                const float* __restrict__ xc,
                const float* __restrict__ A_log,
                const float* __restrict__ Dp,
                float* __restrict__ y) {
    __shared__ float Bs[64][D_STATE];
    __shared__ float Cs[64][D_STATE];

    const int b   = blockIdx.y;
    const int c   = blockIdx.x * 256 + threadIdx.x;
    const int tid = threadIdx.x;

    float Av[D_STATE];
    #pragma unroll
    for (int s = 0; s < D_STATE; ++s) Av[s] = -__expf(A_log[(size_t)c * D_STATE + s]);
    const float Dv = Dp[c];

    float h[D_STATE];
    #pragma unroll
    for (int s = 0; s < D_STATE; ++s) h[s] = 0.0f;

    for (int t0 = 0; t0 < SEQ; t0 += 64) {
        // stage B/C chunk: 64 x 16 floats each; one float4 per thread per matrix
        const int tt = tid >> 2;
        const int s4 = (tid & 3) << 2;
        const size_t rb = ((size_t)(b * SEQ + t0 + tt)) * 96;
        const float4 bv = *(const float4*)&dbc[rb + DT_RANK + s4];
        Bs[tt][s4 + 0] = bv.x; Bs[tt][s4 + 1] = bv.y;
        Bs[tt][s4 + 2] = bv.z; Bs[tt][s4 + 3] = bv.w;
        const float4 cv = *(const float4*)&dbc[rb + DT_RANK + D_STATE + s4];
        Cs[tt][s4 + 0] = cv.x; Cs[tt][s4 + 1] = cv.y;
        Cs[tt][s4 + 2] = cv.z; Cs[tt][s4 + 3] = cv.w;
        __syncthreads();

        for (int t = 0; t < 64; ++t) {
            const size_t off = ((size_t)(b * SEQ + t0 + t)) * D_INNER + c;
            const float d  = delta[off];
            const float u  = xc[off];
            const float du = d * u;
            float acc = Dv * u;
            #pragma unroll
            for (int s = 0; s < D_STATE; ++s) {
                h[s] = __expf(d * Av[s]) * h[s] + du * Bs[t][s];
                acc += h[s] * Cs[t][s];
            }
            y[off] = acc;
        }
        __syncthreads();
    }
}

// ---------------------------------------------------------------------------
// y *= silu(z), z = xz[:, D_INNER:2*D_INNER]
// ---------------------------------------------------------------------------
__global__ void mamba_gate(float* __restrict__ y, const float* __restrict__ xz) {
    const int total = M_ROWS * D_INNER;
    for (int idx = blockIdx.x * blockDim.x + threadIdx.x; idx < total;
         idx += gridDim.x * blockDim.x) {
        const int row = idx >> 11;          // / D_INNER
        const int c   = idx & (D_INNER - 1);
        const float z = xz[(size_t)row * (2 * D_INNER) + D_INNER + c];
        y[idx] = y[idx] * siluf(z);
    }
}

// ---------------------------------------------------------------------------
extern "C" void kernel_launch(void* const* d_in, const int* in_sizes, int n_in,
                              void* d_out, int out_size, void* d_ws, size_t ws_size,
                              hipStream_t stream) {
    const float* x_in   = (const float*)d_in[0];
    const float* norm_w = (const float*)d_in[1];
    const float* in_w   = (const float*)d_in[2];
    const float* conv_w = (const float*)d_in[3];
    const float* conv_b = (const float*)d_in[4];
    const float* xp_w   = (const float*)d_in[5];
    const float* dt_w   = (const float*)d_in[6];
    const float* dt_b   = (const float*)d_in[7];
    const float* A_log  = (const float*)d_in[8];
    const float* D_par  = (const float*)d_in[9];
    const float* out_w  = (const float*)d_in[10];
    float* xout = (float*)d_out;

    // workspace carve-up (floats)
    float* hbuf  = (float*)d_ws;                         // 2048 x 1024
    float* xz    = hbuf  + (size_t)M_ROWS * D_MODEL;     // 2048 x 4096
    float* xc    = xz    + (size_t)M_ROWS * 2 * D_INNER; // 2048 x 2048
    float* dbc   = xc    + (size_t)M_ROWS * D_INNER;     // 2048 x 96
    float* delta = dbc   + (size_t)M_ROWS * 96;          // 2048 x 2048
    float* ybuf  = delta + (size_t)M_ROWS * D_INNER;     // 2048 x 2048

    for (int layer = 0; layer < 2; ++layer) {
        const float* xcur = (layer == 0) ? x_in : (const float*)xout;
        const float* nw  = norm_w + (size_t)layer * D_MODEL;
        const float* iw  = in_w   + (size_t)layer * 2 * D_INNER * D_MODEL;
        const float* cwp = conv_w + (size_t)layer * D_INNER * D_CONV;
        const float* cbp = conv_b + (size_t)layer * D_INNER;
        const float* xpw = xp_w   + (size_t)layer * (DT_RANK + 2 * D_STATE) * D_INNER;
        const float* dtw = dt_w   + (size_t)layer * D_INNER * DT_RANK;
        const float* dtb = dt_b   + (size_t)layer * D_INNER;
        const float* alp = A_log  + (size_t)layer * D_INNER * D_STATE;
        const float* dpp = D_par  + (size_t)layer * D_INNER;
        const float* ow  = out_w  + (size_t)layer * D_MODEL * D_INNER;

        // 1. h = rmsnorm(x) * w
        mamba_rmsnorm<<<M_ROWS, 256, 0, stream>>>(xcur, nw, hbuf, D_MODEL);

        // 2. xz = h @ in_w.T   (2048 x 4096, K=1024)
        mamba_gemm_nt_64x64<<<dim3((2 * D_INNER) / 64, M_ROWS / 64), 256, 0, stream>>>(
            hbuf, iw, nullptr, nullptr, xz,
            D_MODEL, D_MODEL, D_MODEL, 2 * D_INNER, 0);

        // 3. xc = silu(causal_conv(xz[:, :D_INNER]) + cb)
        mamba_conv_silu<<<(BATCH * SEQ * D_INNER) / 256, 256, 0, stream>>>(xz, cwp, cbp, xc);

        // 4. dbc = xc @ xp_w.T  (2048 x 96, K=2048) — skinny N, per-wave GEMM
        {
            const int waves = (M_ROWS / 16) * ((DT_RANK + 2 * D_STATE) / 16); // 768
            mamba_gemm_nt_wave<<<(waves * 32 + 127) / 128, 128, 0, stream>>>(
                xc, xpw, dbc, M_ROWS, DT_RANK + 2 * D_STATE, D_INNER,
                D_INNER, D_INNER, DT_RANK + 2 * D_STATE);
        }

        // 5. delta = softplus(dbc[:, :64] @ dt_w.T + dt_b)  (2048 x 2048, K=64)
        mamba_gemm_nt_64x64<<<dim3(D_INNER / 64, M_ROWS / 64), 256, 0, stream>>>(
            dbc, dtw, dtb, nullptr, delta,
            DT_RANK, 96, DT_RANK, D_INNER, 1);

        // 6. selective scan -> ybuf
        mamba_scan<<<dim3(D_INNER / 256, BATCH), 256, 0, stream>>>(
            dbc, delta, xc, alp, dpp, ybuf);

        // 7. ybuf *= silu(z)
        mamba_gate<<<(M_ROWS * D_INNER) / 256, 256, 0, stream>>>(ybuf, xz);

        // 8. xout = xcur + ybuf @ out_w.T  (2048 x 1024, K=2048), residual fused
        mamba_gemm_nt_64x64<<<dim3(D_MODEL / 64, M_ROWS / 64), 256, 0, stream>>>(
            ybuf, ow, nullptr, xcur, xout,
            D_INNER, D_INNER, D_INNER, D_MODEL, 0);
    }
}